// TRENDSPOT_41437844472184
// MI455X (gfx1250) — compile-verified
//
#include <hip/hip_runtime.h>
#include <hip/hip_bf16.h>
#include <math.h>

typedef float v2f __attribute__((ext_vector_type(2)));
typedef float v8f __attribute__((ext_vector_type(8)));

#define T_STEPS 20
#define IN_DIM_ 8
#define HID_ 64
#define NEG_SLOPE_ 0.2f

// ---------------------------------------------------------------------------
// WMMA helpers (V_WMMA_F32_16X16X4_F32: A 16x4, B 4x16, C/D 16x16 f32)
// A layout: lane holds row m = lane%16; VGPR j holds k = kbase + j + 2*(lane/16)
// B layout (mirror): lane holds col n = lane%16; same k striping
// C/D layout: VGPR v, lane l -> element [m = v + 8*(l/16)][n = l%16]
// ---------------------------------------------------------------------------
__device__ __forceinline__ v8f wmma4(v2f a, v2f b, v8f c) {
  return __builtin_amdgcn_wmma_f32_16x16x4_f32(false, a, false, b, (short)0, c,
                                               false, false);
}

// Load an A/B fragment from an LDS (or any) matrix stored [row][stride],
// where row = rowbase + lane%16 and the k-pair starts at kbase + 2*(lane/16).
__device__ __forceinline__ v2f frag(const float* p0, int stride, int rowbase,
                                    int kbase, int lane) {
  int r = rowbase + (lane & 15);
  int k = kbase + ((lane >> 4) << 1);
  const float* p = p0 + r * stride + k;
  v2f out;
  out.x = p[0];
  out.y = p[1];
  return out;
}

__device__ __forceinline__ v8f splat8(float b) {
  v8f r;
#pragma unroll
  for (int i = 0; i < 8; ++i) r[i] = b;
  return r;
}

__device__ __forceinline__ float sigmoidf_(float x) {
  return 1.0f / (1.0f + __expf(-x));
}

// Monotone float <-> uint map for atomic segment-max over signed floats.
__device__ __forceinline__ unsigned ordF(float f) {
  unsigned u = __float_as_uint(f);
  return (u & 0x80000000u) ? ~u : (u | 0x80000000u);
}
__device__ __forceinline__ float deordF(unsigned u) {
  return (u & 0x80000000u) ? __uint_as_float(u ^ 0x80000000u)
                           : __uint_as_float(~u);
}

// ---------------------------------------------------------------------------
// Kernel 1: fused 2-layer LSTM + temporal attention (online softmax) + W_out
// One wave == 16 nodes. 8 waves / block. All weights LDS-resident.
// ---------------------------------------------------------------------------
__global__ __launch_bounds__(256) void encoder_kernel(
    const float* __restrict__ x, const float* __restrict__ Wih1,
    const float* __restrict__ Whh1, const float* __restrict__ bih1,
    const float* __restrict__ bhh1, const float* __restrict__ Wih2,
    const float* __restrict__ Whh2, const float* __restrict__ bih2,
    const float* __restrict__ bhh2, const float* __restrict__ W_att,
    const float* __restrict__ v_att, const float* __restrict__ W_out,
    const float* __restrict__ b_out, float* __restrict__ x1_out, int Nn) {
  extern __shared__ float lds[];
  float* sWih1 = lds;                  // 256*8
  float* sWhh1 = sWih1 + 256 * 8;      // 256*64
  float* sWih2 = sWhh1 + 256 * 64;     // 256*64
  float* sWhh2 = sWih2 + 256 * 64;     // 256*64
  float* sWattT = sWhh2 + 256 * 64;    // 64*64 (transposed: [n][k])
  float* sWoutT = sWattT + 64 * 64;    // 64*64 (transposed: [n][k])
  float* sB1 = sWoutT + 64 * 64;       // 256 (bih1+bhh1)
  float* sB2 = sB1 + 256;              // 256
  float* sVatt = sB2 + 256;            // 64
  float* sBout = sVatt + 64;           // 64
  float* sbufs = sBout + 64;           // 8 waves * 2048 (h1buf+h2buf)

  int tid = threadIdx.x;
  for (int i = tid; i < 256 * 8; i += 256) sWih1[i] = Wih1[i];
  for (int i = tid; i < 256 * 64; i += 256) {
    sWhh1[i] = Whh1[i];
    sWih2[i] = Wih2[i];
    sWhh2[i] = Whh2[i];
  }
  for (int i = tid; i < 64 * 64; i += 256) {
    int n = i >> 6, k = i & 63;
    sWattT[n * 64 + k] = W_att[k * 64 + n];
    sWoutT[n * 64 + k] = W_out[k * 64 + n];
  }
  sB1[tid] = bih1[tid] + bhh1[tid];
  sB2[tid] = bih2[tid] + bhh2[tid];
  if (tid < 64) {
    sVatt[tid] = v_att[tid];
    sBout[tid] = b_out[tid];
  }

  int wave = tid >> 5, lane = tid & 31;
  float* h1buf = sbufs + wave * 2048;  // [16 nodes][64]
  float* h2buf = h1buf + 1024;         // [16 nodes][64]
  for (int i = lane; i < 2048; i += 32) h1buf[i] = 0.f;
  __syncthreads();

  int nodeBase = (blockIdx.x * 8 + wave) * 16;
  int lhi = (lane >> 4) << 3;  // 0 or 8: m-offset for D layout
  int lcol = lane & 15;

  v8f c1t[4], c2t[4], ctx4[4];
  float rm[8], rd[8];
#pragma unroll
  for (int j = 0; j < 4; ++j) {
    c1t[j] = splat8(0.f);
    c2t[j] = splat8(0.f);
    ctx4[j] = splat8(0.f);
  }
#pragma unroll
  for (int v = 0; v < 8; ++v) {
    rm[v] = -3.0e38f;
    rd[v] = 0.f;
  }

#pragma unroll 1
  for (int t = 0; t < T_STEPS; ++t) {
    // ---------------- LSTM layer 1 ----------------
    v8f nh1[4];
#pragma unroll 1
    for (int jt = 0; jt < 4; ++jt) {
      v8f ai = splat8(sB1[jt * 16 + lcol]);
      v8f af = splat8(sB1[(jt + 4) * 16 + lcol]);
      v8f ag = splat8(sB1[(jt + 8) * 16 + lcol]);
      v8f ao = splat8(sB1[(jt + 12) * 16 + lcol]);
#pragma unroll
      for (int kk = 0; kk < 2; ++kk) {  // x @ Wih1^T, K=8
        int m = nodeBase + lcol;
        if (m >= Nn) m = Nn - 1;
        int k = kk * 4 + ((lane >> 4) << 1);
        const float* px = x + ((size_t)m * T_STEPS + t) * IN_DIM_ + k;
        v2f a;
        a.x = px[0];
        a.y = px[1];
        ai = wmma4(a, frag(sWih1, 8, (jt) * 16, kk * 4, lane), ai);
        af = wmma4(a, frag(sWih1, 8, (jt + 4) * 16, kk * 4, lane), af);
        ag = wmma4(a, frag(sWih1, 8, (jt + 8) * 16, kk * 4, lane), ag);
        ao = wmma4(a, frag(sWih1, 8, (jt + 12) * 16, kk * 4, lane), ao);
      }
#pragma unroll
      for (int kk = 0; kk < 16; ++kk) {  // h1 @ Whh1^T, K=64
        v2f a = frag(h1buf, 64, 0, kk * 4, lane);
        ai = wmma4(a, frag(sWhh1, 64, (jt) * 16, kk * 4, lane), ai);
        af = wmma4(a, frag(sWhh1, 64, (jt + 4) * 16, kk * 4, lane), af);
        ag = wmma4(a, frag(sWhh1, 64, (jt + 8) * 16, kk * 4, lane), ag);
        ao = wmma4(a, frag(sWhh1, 64, (jt + 12) * 16, kk * 4, lane), ao);
      }
#pragma unroll
      for (int v = 0; v < 8; ++v) {
        float c = sigmoidf_(af[v]) * c1t[jt][v] + sigmoidf_(ai[v]) * tanhf(ag[v]);
        c1t[jt][v] = c;
        nh1[jt][v] = sigmoidf_(ao[v]) * tanhf(c);
      }
    }
#pragma unroll
    for (int jt = 0; jt < 4; ++jt)
#pragma unroll
      for (int v = 0; v < 8; ++v)
        h1buf[(v + lhi) * 64 + jt * 16 + lcol] = nh1[jt][v];

    // ---------------- LSTM layer 2 ----------------
    v8f nh2[4];
#pragma unroll 1
    for (int jt = 0; jt < 4; ++jt) {
      v8f ai = splat8(sB2[jt * 16 + lcol]);
      v8f af = splat8(sB2[(jt + 4) * 16 + lcol]);
      v8f ag = splat8(sB2[(jt + 8) * 16 + lcol]);
      v8f ao = splat8(sB2[(jt + 12) * 16 + lcol]);
#pragma unroll
      for (int kk = 0; kk < 16; ++kk) {  // h1(new) @ Wih2^T
        v2f a = frag(h1buf, 64, 0, kk * 4, lane);
        ai = wmma4(a, frag(sWih2, 64, (jt) * 16, kk * 4, lane), ai);
        af = wmma4(a, frag(sWih2, 64, (jt + 4) * 16, kk * 4, lane), af);
        ag = wmma4(a, frag(sWih2, 64, (jt + 8) * 16, kk * 4, lane), ag);
        ao = wmma4(a, frag(sWih2, 64, (jt + 12) * 16, kk * 4, lane), ao);
      }
#pragma unroll
      for (int kk = 0; kk < 16; ++kk) {  // h2(prev) @ Whh2^T
        v2f a = frag(h2buf, 64, 0, kk * 4, lane);
        ai = wmma4(a, frag(sWhh2, 64, (jt) * 16, kk * 4, lane), ai);
        af = wmma4(a, frag(sWhh2, 64, (jt + 4) * 16, kk * 4, lane), af);
        ag = wmma4(a, frag(sWhh2, 64, (jt + 8) * 16, kk * 4, lane), ag);
        ao = wmma4(a, frag(sWhh2, 64, (jt + 12) * 16, kk * 4, lane), ao);
      }
#pragma unroll
      for (int v = 0; v < 8; ++v) {
        float c = sigmoidf_(af[v]) * c2t[jt][v] + sigmoidf_(ai[v]) * tanhf(ag[v]);
        c2t[jt][v] = c;
        nh2[jt][v] = sigmoidf_(ao[v]) * tanhf(c);
      }
    }
#pragma unroll
    for (int jt = 0; jt < 4; ++jt)
#pragma unroll
      for (int v = 0; v < 8; ++v)
        h2buf[(v + lhi) * 64 + jt * 16 + lcol] = nh2[jt][v];

    // ---------- attention score + online softmax-weighted sum ----------
    float pv[8];
#pragma unroll
    for (int v = 0; v < 8; ++v) pv[v] = 0.f;
#pragma unroll 1
    for (int nt = 0; nt < 4; ++nt) {
      v8f u = splat8(0.f);
#pragma unroll
      for (int kk = 0; kk < 16; ++kk)
        u = wmma4(frag(h2buf, 64, 0, kk * 4, lane),
                  frag(sWattT, 64, nt * 16, kk * 4, lane), u);
      float va = sVatt[nt * 16 + lcol];
#pragma unroll
      for (int v = 0; v < 8; ++v) pv[v] += tanhf(u[v]) * va;
    }
#pragma unroll
    for (int off = 1; off < 16; off <<= 1) {
#pragma unroll
      for (int v = 0; v < 8; ++v) pv[v] += __shfl_xor(pv[v], off, 32);
    }
#pragma unroll
    for (int v = 0; v < 8; ++v) {
      float s = pv[v];
      float nm = fmaxf(rm[v], s);
      float al = __expf(rm[v] - nm);
      float w = __expf(s - nm);
      rd[v] = rd[v] * al + w;
      rm[v] = nm;
#pragma unroll
      for (int jt = 0; jt < 4; ++jt)
        ctx4[jt][v] = ctx4[jt][v] * al + w * nh2[jt][v];
    }
  }  // t

  // normalize ctx, project with W_out, store x1
#pragma unroll
  for (int v = 0; v < 8; ++v) {
    float inv = 1.0f / rd[v];
#pragma unroll
    for (int jt = 0; jt < 4; ++jt) ctx4[jt][v] *= inv;
  }
#pragma unroll
  for (int jt = 0; jt < 4; ++jt)
#pragma unroll
    for (int v = 0; v < 8; ++v)
      h1buf[(v + lhi) * 64 + jt * 16 + lcol] = ctx4[jt][v];

#pragma unroll 1
  for (int nt = 0; nt < 4; ++nt) {
    v8f a = splat8(sBout[nt * 16 + lcol]);
#pragma unroll
    for (int kk = 0; kk < 16; ++kk)
      a = wmma4(frag(h1buf, 64, 0, kk * 4, lane),
                frag(sWoutT, 64, nt * 16, kk * 4, lane), a);
#pragma unroll
    for (int v = 0; v < 8; ++v) {
      int m = nodeBase + v + lhi;
      if (m < Nn) x1_out[(size_t)m * 64 + nt * 16 + lcol] = a[v];
    }
  }
}

// ---------------------------------------------------------------------------
// Kernel 2: GAT node transform: h = (xin [+bias]) @ Wg ; as/ad = h . a_src/dst
// ---------------------------------------------------------------------------
__global__ __launch_bounds__(256) void gat_xform_kernel(
    const float* __restrict__ xin, const float* __restrict__ in_bias, int inDim,
    const float* __restrict__ Wg, const float* __restrict__ a_src,
    const float* __restrict__ a_dst, float* __restrict__ h_out,
    float* __restrict__ as_out, float* __restrict__ ad_out, int Nn) {
  extern __shared__ float lds[];
  float* sWT = lds;  // [32][inDim] = Wg transposed
  float* sAs = sWT + 32 * inDim;
  float* sAd = sAs + 32;
  int tid = threadIdx.x;
  for (int i = tid; i < 32 * inDim; i += 256) {
    int n = i / inDim, k = i - n * inDim;
    sWT[i] = Wg[k * 32 + n];
  }
  if (tid < 32) {
    sAs[tid] = a_src[tid];
    sAd[tid] = a_dst[tid];
  }
  __syncthreads();

  int wave = tid >> 5, lane = tid & 31;
  int nodeBase = (blockIdx.x * 8 + wave) * 16;
  int lhi = (lane >> 4) << 3;
  int lcol = lane & 15;
  int ksteps = inDim >> 2;

  float pa[8], pd[8];
#pragma unroll
  for (int v = 0; v < 8; ++v) {
    pa[v] = 0.f;
    pd[v] = 0.f;
  }
#pragma unroll 1
  for (int nt = 0; nt < 2; ++nt) {
    v8f acc = splat8(0.f);
    for (int kk = 0; kk < ksteps; ++kk) {
      int m = nodeBase + lcol;
      if (m >= Nn) m = Nn - 1;
      int k = kk * 4 + ((lane >> 4) << 1);
      const float* p = xin + (size_t)m * inDim + k;
      v2f a;
      a.x = p[0];
      a.y = p[1];
      if (in_bias) {
        a.x += in_bias[k];
        a.y += in_bias[k + 1];
      }
      acc = wmma4(a, frag(sWT, inDim, nt * 16, kk * 4, lane), acc);
    }
    float vs = sAs[nt * 16 + lcol];
    float vd = sAd[nt * 16 + lcol];
#pragma unroll
    for (int v = 0; v < 8; ++v) {
      int m = nodeBase + v + lhi;
      if (m < Nn) h_out[(size_t)m * 32 + nt * 16 + lcol] = acc[v];
      pa[v] += acc[v] * vs;
      pd[v] += acc[v] * vd;
    }
  }
#pragma unroll
  for (int off = 1; off < 16; off <<= 1) {
#pragma unroll
    for (int v = 0; v < 8; ++v) {
      pa[v] += __shfl_xor(pa[v], off, 32);
      pd[v] += __shfl_xor(pd[v], off, 32);
    }
  }
  if (lcol == 0) {
#pragma unroll
    for (int v = 0; v < 8; ++v) {
      int m = nodeBase + v + lhi;
      if (m < Nn) {
        as_out[m] = pa[v];
        ad_out[m] = pd[v];
      }
    }
  }
}

// ---------------------------------------------------------------------------
// Edge kernels (3 passes): segment-max, segment-sum(exp), weighted aggregate
// ---------------------------------------------------------------------------
__device__ __forceinline__ bool edge_sd(const int* ei, long long e, int E,
                                        int Nn, int& s, int& d) {
  if (e >= (long long)E + Nn) return false;
  if (e < E) {
    s = ei[e];
    d = ei[(size_t)E + e];
  } else {
    s = d = (int)(e - E);
  }
  return true;
}

__global__ void edge_max_kernel(const int* __restrict__ ei, int E, int Nn,
                                const float* __restrict__ as_,
                                const float* __restrict__ ad_,
                                unsigned* __restrict__ emax) {
  long long e = (long long)blockIdx.x * blockDim.x + threadIdx.x;
  int s, d;
  if (!edge_sd(ei, e, E, Nn, s, d)) return;
  float v = as_[s] + ad_[d];
  v = v > 0.f ? v : NEG_SLOPE_ * v;
  atomicMax(&emax[d], ordF(v));
}

__global__ void edge_sum_kernel(const int* __restrict__ ei, int E, int Nn,
                                const float* __restrict__ as_,
                                const float* __restrict__ ad_,
                                const unsigned* __restrict__ emax,
                                float* __restrict__ denom) {
  long long e = (long long)blockIdx.x * blockDim.x + threadIdx.x;
  int s, d;
  if (!edge_sd(ei, e, E, Nn, s, d)) return;
  float v = as_[s] + ad_[d];
  v = v > 0.f ? v : NEG_SLOPE_ * v;
  float ex = __expf(v - deordF(emax[d]));
  atomicAdd(&denom[d], ex);
}

__global__ void edge_agg_kernel(const int* __restrict__ ei, int E, int Nn,
                                const float* __restrict__ as_,
                                const float* __restrict__ ad_,
                                const unsigned* __restrict__ emax,
                                const float* __restrict__ denom,
                                const float* __restrict__ h,
                                float* __restrict__ outp) {
  long long e = (long long)blockIdx.x * blockDim.x + threadIdx.x;
  int s, d;
  if (!edge_sd(ei, e, E, Nn, s, d)) return;
  float v = as_[s] + ad_[d];
  v = v > 0.f ? v : NEG_SLOPE_ * v;
  float alpha = __expf(v - deordF(emax[d])) / denom[d];
  const float* hs = h + (size_t)s * 32;
  float* od = outp + (size_t)d * 32;
#pragma unroll
  for (int c = 0; c < 32; c += 4) {
    float4 hv = *(const float4*)(hs + c);
    atomicAdd(od + c + 0, hv.x * alpha);
    atomicAdd(od + c + 1, hv.y * alpha);
    atomicAdd(od + c + 2, hv.z * alpha);
    atomicAdd(od + c + 3, hv.w * alpha);
  }
}

// ---------------------------------------------------------------------------
// Final head: x3 = relu((out2 + b_g2) @ W_lin + b_lin); out = [log_softmax, x3]
// ---------------------------------------------------------------------------
__global__ void head_kernel(const float* __restrict__ o2,
                            const float* __restrict__ b_g2,
                            const float* __restrict__ W_lin,
                            const float* __restrict__ b_lin,
                            float* __restrict__ out, int Nn) {
  int n = blockIdx.x * blockDim.x + threadIdx.x;
  if (n >= Nn) return;
  float y0 = b_lin[0], y1 = b_lin[1], y2 = b_lin[2];
  const float* p = o2 + (size_t)n * 32;
#pragma unroll
  for (int c = 0; c < 32; ++c) {
    float w = p[c] + b_g2[c];
    y0 += w * W_lin[c * 3 + 0];
    y1 += w * W_lin[c * 3 + 1];
    y2 += w * W_lin[c * 3 + 2];
  }
  y0 = fmaxf(y0, 0.f);
  y1 = fmaxf(y1, 0.f);
  y2 = fmaxf(y2, 0.f);
  float m = fmaxf(y0, fmaxf(y1, y2));
  float lse = logf(__expf(y0 - m) + __expf(y1 - m) + __expf(y2 - m)) + m;
  out[(size_t)n * 3 + 0] = y0 - lse;
  out[(size_t)n * 3 + 1] = y1 - lse;
  out[(size_t)n * 3 + 2] = y2 - lse;
  float* x3 = out + (size_t)Nn * 3;
  x3[(size_t)n * 3 + 0] = y0;
  x3[(size_t)n * 3 + 1] = y1;
  x3[(size_t)n * 3 + 2] = y2;
}

// ---------------------------------------------------------------------------
extern "C" void kernel_launch(void* const* d_in, const int* in_sizes, int n_in,
                              void* d_out, int out_size, void* d_ws,
                              size_t ws_size, hipStream_t stream) {
  const float* x = (const float*)d_in[0];
  const int* ei = (const int*)d_in[1];
  const float* Wih1 = (const float*)d_in[2];
  const float* Whh1 = (const float*)d_in[3];
  const float* bih1 = (const float*)d_in[4];
  const float* bhh1 = (const float*)d_in[5];
  const float* Wih2 = (const float*)d_in[6];
  const float* Whh2 = (const float*)d_in[7];
  const float* bih2 = (const float*)d_in[8];
  const float* bhh2 = (const float*)d_in[9];
  const float* W_att = (const float*)d_in[10];
  const float* v_att = (const float*)d_in[11];
  const float* W_out = (const float*)d_in[12];
  const float* b_out = (const float*)d_in[13];
  const float* W_g1 = (const float*)d_in[14];
  const float* a_src1 = (const float*)d_in[15];
  const float* a_dst1 = (const float*)d_in[16];
  const float* b_g1 = (const float*)d_in[17];
  const float* W_g2 = (const float*)d_in[18];
  const float* a_src2 = (const float*)d_in[19];
  const float* a_dst2 = (const float*)d_in[20];
  const float* b_g2 = (const float*)d_in[21];
  const float* W_lin = (const float*)d_in[22];
  const float* b_lin = (const float*)d_in[23];

  int N = in_sizes[0] / (T_STEPS * IN_DIM_);
  int E = in_sizes[1] / 2;
  size_t Ns = (size_t)N;

  // workspace layout (floats)
  float* ws = (float*)d_ws;
  float* x1 = ws;                      // N*64
  float* h1g = x1 + Ns * 64;           // N*32
  float* as1 = h1g + Ns * 32;          // N
  float* ad1 = as1 + Ns;               // N
  float* den1 = ad1 + Ns;              // N   (zeroed)
  unsigned* em1 = (unsigned*)(den1 + Ns);  // N  (zeroed == ord(-inf))
  float* o1 = (float*)(em1 + Ns);      // N*32 (zeroed)
  float* h2g = o1 + Ns * 32;           // N*32
  float* as2 = h2g + Ns * 32;          // N
  float* ad2 = as2 + Ns;               // N
  float* den2 = ad2 + Ns;              // N   (zeroed)
  unsigned* em2 = (unsigned*)(den2 + Ns);  // N (zeroed)
  float* o2 = (float*)(em2 + Ns);      // N*32 (zeroed)

  hipMemsetAsync(den1, 0, (size_t)34 * Ns * sizeof(float), stream);
  hipMemsetAsync(den2, 0, (size_t)34 * Ns * sizeof(float), stream);

  int nodeBlocks = (N + 127) / 128;
  size_t ldsEnc =
      (size_t)(256 * 8 + 3 * 256 * 64 + 2 * 64 * 64 + 2 * 256 + 2 * 64 +
               8 * 2048) *
      sizeof(float);
  encoder_kernel<<<nodeBlocks, 256, ldsEnc, stream>>>(
      x, Wih1, Whh1, bih1, bhh1, Wih2, Whh2, bih2, bhh2, W_att, v_att, W_out,
      b_out, x1, N);

  size_t ldsG1 = (size_t)(32 * 64 + 64) * sizeof(float);
  gat_xform_kernel<<<nodeBlocks, 256, ldsG1, stream>>>(
      x1, nullptr, 64, W_g1, a_src1, a_dst1, h1g, as1, ad1, N);

  long long ETot = (long long)E + N;
  int edgeBlocks = (int)((ETot + 255) / 256);
  edge_max_kernel<<<edgeBlocks, 256, 0, stream>>>(ei, E, N, as1, ad1, em1);
  edge_sum_kernel<<<edgeBlocks, 256, 0, stream>>>(ei, E, N, as1, ad1, em1,
                                                  den1);
  edge_agg_kernel<<<edgeBlocks, 256, 0, stream>>>(ei, E, N, as1, ad1, em1,
                                                  den1, h1g, o1);

  size_t ldsG2 = (size_t)(32 * 32 + 64) * sizeof(float);
  gat_xform_kernel<<<nodeBlocks, 256, ldsG2, stream>>>(
      o1, b_g1, 32, W_g2, a_src2, a_dst2, h2g, as2, ad2, N);
  edge_max_kernel<<<edgeBlocks, 256, 0, stream>>>(ei, E, N, as2, ad2, em2);
  edge_sum_kernel<<<edgeBlocks, 256, 0, stream>>>(ei, E, N, as2, ad2, em2,
                                                  den2);
  edge_agg_kernel<<<edgeBlocks, 256, 0, stream>>>(ei, E, N, as2, ad2, em2,
                                                  den2, h2g, o2);

  head_kernel<<<(N + 255) / 256, 256, 0, stream>>>(o2, b_g2, W_lin, b_lin,
                                                   (float*)d_out, N);
}